// SELayer_19164144075614
// MI455X (gfx1250) — compile-verified
//
#include <hip/hip_runtime.h>
#include <math.h>

typedef __attribute__((ext_vector_type(2))) float v2f;
typedef __attribute__((ext_vector_type(4))) float v4f;
typedef __attribute__((ext_vector_type(8))) float v8f;

#define SE_C  512
#define SE_CR 32
#define SE_L  4096
#define SE_B  64
#define NROWS (SE_B * SE_C)        // 32768 (b,c) rows

// ---------------------------------------------------------------------------
// Kernel 1: mean over L for each (b,c) row. One wave32 per row.
// Each lane streams 32 x float4 (b128) non-temporal loads, then shfl-xor tree.
// ---------------------------------------------------------------------------
__global__ void se_mean_kernel(const float* __restrict__ x,
                               float* __restrict__ mean) {
  const int wave = threadIdx.x >> 5;          // 8 waves per block
  const int lane = threadIdx.x & 31;
  const int row  = blockIdx.x * 8 + wave;     // 0..32767
  const v4f* xr = (const v4f*)(x + (size_t)row * SE_L);

  float s = 0.f;
#pragma unroll 8
  for (int it = 0; it < (SE_L / 4) / 32; ++it) {   // 32 iters
    v4f v = __builtin_nontemporal_load(&xr[lane + 32 * it]);
    s += v.x + v.y + v.z + v.w;
  }
#pragma unroll
  for (int off = 16; off > 0; off >>= 1)
    s += __shfl_xor(s, off, 32);
  if (lane == 0) mean[row] = s * (1.0f / SE_L);
}

// ---------------------------------------------------------------------------
// Kernel 2: excitation MLP, single workgroup (256 threads = 8 waves), via
// V_WMMA_F32_16X16X4_F32 tiles.
//   Layer 1: [64,512] @ w1^T[512,32] + b1, ReLU -> LDS y1[64][32] (stride 33)
//   Layer 2: [64,32]  @ w2^T[32,512] + b2, sigmoid -> scale[64,512]
// Fragment layouts per CDNA5 ISA 7.12.2 (32-bit A 16x4, 32-bit C/D 16x16).
// ---------------------------------------------------------------------------
__global__ void se_mlp_kernel(const float* __restrict__ mean,
                              const float* __restrict__ w1,
                              const float* __restrict__ b1,
                              const float* __restrict__ w2,
                              const float* __restrict__ b2,
                              float* __restrict__ scale) {
  __shared__ float y1[SE_B * 33];             // [64][32], padded to 33

  const int wave = threadIdx.x >> 5;          // 0..7
  const int lane = threadIdx.x & 31;
  const int half = lane >> 4;                 // 0 or 1
  const int ln16 = lane & 15;

  // ---- Layer 1: M=64, N=32, K=512 -> 4x2 = 8 tiles, one per wave
  {
    const int mt = wave >> 1;                 // 0..3
    const int nt = wave & 1;                  // 0..1
    const int m  = mt * 16 + ln16;            // A row for this lane
    const int n  = nt * 16 + ln16;            // B col / D col for this lane
    v8f acc = {};
    for (int k0 = 0; k0 < SE_C; k0 += 4) {
      v2f a, b;
      // A = mean [64 x 512]: a.x = A[m][k0+2*half], a.y = A[m][k0+2*half+1]
      a.x = mean[m * SE_C + k0 + 2 * half];
      a.y = mean[m * SE_C + k0 + 2 * half + 1];
      // B = w1^T [512 x 32]: B[k][n] = w1[n*512 + k]  (w1 is [32,512])
      b.x = w1[n * SE_C + k0 + 2 * half];
      b.y = w1[n * SE_C + k0 + 2 * half + 1];
      acc = __builtin_amdgcn_wmma_f32_16x16x4_f32(
          false, a, false, b, (short)0, acc, false, false);
    }
    const float bias = b1[n];
#pragma unroll
    for (int j = 0; j < 8; ++j) {
      const int mrow = mt * 16 + j + 8 * half;     // D: vgpr j, half selects +8
      float v = acc[j] + bias;
      y1[mrow * 33 + n] = v > 0.f ? v : 0.f;       // ReLU
    }
  }
  __syncthreads();

  // ---- Layer 2: M=64, N=512, K=32 -> 4x32 = 128 tiles, 16 per wave
  for (int t = wave * 16; t < wave * 16 + 16; ++t) {
    const int mt = t >> 5;                    // 0..3
    const int nt = t & 31;                    // 0..31
    const int m  = mt * 16 + ln16;
    const int n  = nt * 16 + ln16;
    v8f acc = {};
#pragma unroll
    for (int k0 = 0; k0 < SE_CR; k0 += 4) {
      v2f a, b;
      // A = y1 [64 x 32] in LDS (stride 33)
      a.x = y1[m * 33 + k0 + 2 * half];
      a.y = y1[m * 33 + k0 + 2 * half + 1];
      // B = w2^T [32 x 512]: B[k][n] = w2[n*32 + k]  (w2 is [512,32])
      b.x = w2[n * SE_CR + k0 + 2 * half];
      b.y = w2[n * SE_CR + k0 + 2 * half + 1];
      acc = __builtin_amdgcn_wmma_f32_16x16x4_f32(
          false, a, false, b, (short)0, acc, false, false);
    }
    const float bias = b2[n];
#pragma unroll
    for (int j = 0; j < 8; ++j) {
      const int mrow = mt * 16 + j + 8 * half;
      float v = acc[j] + bias;
      scale[mrow * SE_C + n] = 1.0f / (1.0f + __expf(-v));   // sigmoid
    }
  }
}

// ---------------------------------------------------------------------------
// Kernel 3: out[b,c,l] = x[b,c,l] * scale[b,c]. Pure streaming, float4.
// ---------------------------------------------------------------------------
__global__ void se_scale_kernel(const float* __restrict__ x,
                                const float* __restrict__ scale,
                                float* __restrict__ out) {
  const size_t N4 = (size_t)SE_B * SE_C * SE_L / 4;   // 33,554,432
  const v4f* x4 = (const v4f*)x;
  v4f* o4 = (v4f*)out;
  size_t idx = (size_t)blockIdx.x * blockDim.x + threadIdx.x;
  const size_t stride = (size_t)gridDim.x * blockDim.x;
  for (; idx < N4; idx += stride) {
    const float s = scale[idx >> 10];         // 4096/4 = 1024 float4 per row
    v4f v = __builtin_nontemporal_load(&x4[idx]);
    __builtin_nontemporal_store(v * s, &o4[idx]);
  }
}

// ---------------------------------------------------------------------------
extern "C" void kernel_launch(void* const* d_in, const int* in_sizes, int n_in,
                              void* d_out, int out_size, void* d_ws, size_t ws_size,
                              hipStream_t stream) {
  const float* x  = (const float*)d_in[0];
  const float* w1 = (const float*)d_in[1];
  const float* b1 = (const float*)d_in[2];
  const float* w2 = (const float*)d_in[3];
  const float* b2 = (const float*)d_in[4];
  float* out = (float*)d_out;

  float* mean  = (float*)d_ws;                // 32768 floats (128 KB)
  float* scale = (float*)d_ws + NROWS;        // 32768 floats (128 KB)

  // 1) squeeze: 4096 blocks x 8 waves = 32768 rows
  se_mean_kernel<<<NROWS / 8, 256, 0, stream>>>(x, mean);
  // 2) excitation MLP: single workgroup, WMMA path
  se_mlp_kernel<<<1, 256, 0, stream>>>(mean, w1, b1, w2, b2, scale);
  // 3) broadcast scale: grid-stride float4 stream
  se_scale_kernel<<<65536, 256, 0, stream>>>(x, scale, out);
}